// P_GAT_2413771621183
// MI455X (gfx1250) — compile-verified
//
#include <hip/hip_runtime.h>
#include <hip/hip_bf16.h>

typedef __attribute__((ext_vector_type(2))) float v2f;
typedef __attribute__((ext_vector_type(8))) float v8f;

#define BDIM  16
#define PDIM  1024
#define DDIM  256
#define HDIM  4
#define DH    64

// ---------------------------------------------------------------------------
// GEMM1: qk = x @ W^T.  A = x (M=B*P rows, K=256), B[k][n] = W[n][k].
// Block = 128 threads (4 waves). Each wave computes one 16x16 tile with
// V_WMMA_F32_16X16X4_F32 over a K=256 loop. grid = (M/16, N/64).
// ---------------------------------------------------------------------------
__global__ void gemm_qk_kernel(const float* __restrict__ x,
                               const float* __restrict__ W,
                               float* __restrict__ qk) {
    const int wave = threadIdx.x >> 5;
    const int lane = threadIdx.x & 31;
    const int m0 = blockIdx.x * 16;
    const int n0 = blockIdx.y * 64 + wave * 16;

    const int m    = lane & 15;   // A-fragment row
    const int n    = lane & 15;   // B-fragment col
    const int half = lane >> 4;   // K-half selector (lanes 16-31 hold K+2,K+3)

    const float* __restrict__ arow = x + (size_t)(m0 + m) * DDIM;
    const float* __restrict__ brow = W + (size_t)(n0 + n) * DDIM; // W row n => B column n

    v8f c = {};
#pragma unroll 4
    for (int k = 0; k < DDIM; k += 4) {
        v2f a, b;
        a.x = arow[k + 2 * half];
        a.y = arow[k + 2 * half + 1];
        b.x = brow[k + 2 * half];
        b.y = brow[k + 2 * half + 1];
        c = __builtin_amdgcn_wmma_f32_16x16x4_f32(
                false, a, false, b, (short)0, c, false, false);
    }

    float* __restrict__ drow = qk + (size_t)m0 * DDIM + n0 + n;
#pragma unroll
    for (int v = 0; v < 8; ++v) {
        drow[(size_t)(v + 8 * half) * DDIM] = c[v];
    }
}

// ---------------------------------------------------------------------------
// aq[b,p,h] = dot(qk[b,p,h*64:...], wa_q[h]);  ak likewise with wa_k.
// attn_w layout: (H, 2*DH) => row h: [0:64)=wa_q, [64:128)=wa_k.
// ---------------------------------------------------------------------------
__global__ void aqak_kernel(const float* __restrict__ qk,
                            const float* __restrict__ attn_w,
                            float* __restrict__ aq,
                            float* __restrict__ ak) {
    int idx = blockIdx.x * blockDim.x + threadIdx.x;   // [bp][h]
    if (idx >= BDIM * PDIM * HDIM) return;
    int h  = idx & (HDIM - 1);
    int bp = idx >> 2;
    const float* __restrict__ v  = qk + (size_t)bp * DDIM + h * DH;
    const float* __restrict__ wq = attn_w + h * (2 * DH);
    const float* __restrict__ wk = wq + DH;
    float sq = 0.f, sk = 0.f;
#pragma unroll 8
    for (int d = 0; d < DH; ++d) {
        float t = v[d];
        sq += t * wq[d];
        sk += t * wk[d];
    }
    aq[idx] = sq;
    ak[idx] = sk;
}

// ---------------------------------------------------------------------------
// Attention + output update:
//   w[q,k]    = exp( mask ? -1e-8 : tanh(aq[q]+ak[k]) )   (scores in [-1,1],
//               so exp without max-shift is safe & softmax-equivalent)
//   numer     = w @ qk_head   (K=1024, N=DH=64)
//   x[b,q,:] += tanh(numer / rowsum(w))
// Block = 128 threads (4 waves), one block per (b,h, 16-row q-tile).
// w is built 16x32 at a time in LDS; denom accumulated via ds_add_f32.
// ---------------------------------------------------------------------------
__global__ void attn_kernel(const float* __restrict__ qk,
                            const float* __restrict__ aq,
                            const float* __restrict__ ak,
                            const float* __restrict__ p_mask,
                            float* __restrict__ x) {
    __shared__ float wtile[16][33];   // +1 pad to stagger banks
    __shared__ float denom[16];
    __shared__ float aqs[16];
    __shared__ float mqs[16];

    const int bh = blockIdx.x;
    const int b  = bh >> 2;
    const int h  = bh & (HDIM - 1);
    const int q0 = blockIdx.y * 16;

    const int tid  = threadIdx.x;
    const int wave = tid >> 5;
    const int lane = tid & 31;
    const int m    = lane & 15;
    const int n    = lane & 15;
    const int half = lane >> 4;
    const int n0   = wave * 16;

    if (tid < 16) {
        denom[tid] = 0.f;
        aqs[tid]   = aq[(size_t)(b * PDIM + q0 + tid) * HDIM + h];
        mqs[tid]   = p_mask[b * PDIM + q0 + tid];
    }
    __syncthreads();

    const int wq = tid >> 3;           // q row this thread fills (0..15)
    const int wk = (tid & 7) * 4;      // 4 consecutive k slots
    const float aqv = aqs[wq];
    const float mq  = mqs[wq];

    v8f c = {};
    for (int k0 = 0; k0 < PDIM; k0 += 32) {
        // --- build 16x32 exp(tanh) tile + row sums ---
        float s = 0.f;
#pragma unroll
        for (int j = 0; j < 4; ++j) {
            const int k = k0 + wk + j;
            const float mk = p_mask[b * PDIM + k];
            float sc = tanhf(aqv + ak[(size_t)(b * PDIM + k) * HDIM + h]);
            sc = (mq * mk == 0.f) ? -1e-8f : sc;
            const float wv = __expf(sc);
            wtile[wq][wk + j] = wv;
            s += wv;
        }
        atomicAdd(&denom[wq], s);      // ds_add_f32
        __syncthreads();

        // --- 8 fp32 WMMAs against qk head slice ---
#pragma unroll
        for (int kk = 0; kk < 32; kk += 4) {
            v2f a, bf;
            a.x = wtile[m][kk + 2 * half];
            a.y = wtile[m][kk + 2 * half + 1];
            const float* __restrict__ bp =
                qk + (size_t)(b * PDIM + k0 + kk + 2 * half) * DDIM + h * DH + n0 + n;
            bf.x = bp[0];
            bf.y = bp[DDIM];
            c = __builtin_amdgcn_wmma_f32_16x16x4_f32(
                    false, a, false, bf, (short)0, c, false, false);
        }
        __syncthreads();               // wtile reused next iteration
    }

    // --- epilogue: hid = tanh(numer/denom); x += hid ---
    float* __restrict__ out = x + (size_t)(b * PDIM + q0) * DDIM + h * DH + n0 + n;
#pragma unroll
    for (int v = 0; v < 8; ++v) {
        const int mm = v + 8 * half;
        const float hid = tanhf(c[v] / denom[mm]);
        out[(size_t)mm * DDIM] += hid;
    }
}

// ---------------------------------------------------------------------------
// Launch: x <- feature; then per layer: gemm_qk, aqak, attn (x updated inplace)
// ---------------------------------------------------------------------------
extern "C" void kernel_launch(void* const* d_in, const int* in_sizes, int n_in,
                              void* d_out, int out_size, void* d_ws, size_t ws_size,
                              hipStream_t stream) {
    const float* p_mask  = (const float*)d_in[0];
    const float* feature = (const float*)d_in[1];
    const float* W       = (const float*)d_in[2];
    const float* attn_w  = (const float*)d_in[3];
    (void)in_sizes; (void)n_in; (void)out_size; (void)ws_size;

    float* x = (float*)d_out;                       // running features (B,P,D)

    char* ws = (char*)d_ws;
    float* qk = (float*)ws;                          // B*P*D floats = 16 MB
    float* aq = (float*)(ws + (size_t)BDIM * PDIM * DDIM * sizeof(float));
    float* ak = aq + (size_t)BDIM * PDIM * HDIM;     // each 256 KB

    // x = feature
    hipMemcpyAsync(x, feature, (size_t)BDIM * PDIM * DDIM * sizeof(float),
                   hipMemcpyDeviceToDevice, stream);

    const dim3 gemm_grid(BDIM * PDIM / 16, DDIM / 64);
    const dim3 gemm_blk(128);
    const int  aqak_threads = BDIM * PDIM * HDIM;
    const dim3 attn_grid(BDIM * HDIM, PDIM / 16);
    const dim3 attn_blk(128);

    for (int layer = 0; layer < 2; ++layer) {
        gemm_qk_kernel<<<gemm_grid, gemm_blk, 0, stream>>>(x, W, qk);
        aqak_kernel<<<(aqak_threads + 255) / 256, 256, 0, stream>>>(qk, attn_w, aq, ak);
        attn_kernel<<<attn_grid, attn_blk, 0, stream>>>(qk, aq, ak, p_mask, x);
    }
}